// SGC_33208687133424
// MI455X (gfx1250) — compile-verified
//
#include <hip/hip_runtime.h>
#include <hip/hip_bf16.h>

typedef __attribute__((ext_vector_type(2))) float v2f;
typedef __attribute__((ext_vector_type(8))) float v8f;

#define F_IN  128
#define F_OUT 40

// ---------------------------------------------------------------------------
// Hop kernel: one wave (32 lanes) per edge. Each lane moves a float4 slice of
// the 128-float feature row: gather from h_in[src], atomic-add into h_out[dst].
// edge_mask is {0,1}: mask==0 edges contribute nothing, so skip the wave.
// Lowers to global_load_b128 + 4x global_atomic_add_f32 per lane.
// ---------------------------------------------------------------------------
__global__ __launch_bounds__(256) void sgc_hop_kernel(
    const float* __restrict__ h_in,
    float* __restrict__ h_out,
    const long long* __restrict__ src,
    const long long* __restrict__ dst,
    const int* __restrict__ mask,
    int E)
{
    int gtid = blockIdx.x * blockDim.x + threadIdx.x;
    int edge = gtid >> 5;
    int lane = gtid & 31;
    if (edge >= E) return;
    if (mask[edge] == 0) return;

    long long s = src[edge];
    long long d = dst[edge];

    const float4* in4 = (const float4*)(h_in + (size_t)s * F_IN);
    float4 v = in4[lane];

    float* outp = h_out + (size_t)d * F_IN + (size_t)lane * 4;
    unsafeAtomicAdd(outp + 0, v.x);
    unsafeAtomicAdd(outp + 1, v.y);
    unsafeAtomicAdd(outp + 2, v.z);
    unsafeAtomicAdd(outp + 3, v.w);
}

// ---------------------------------------------------------------------------
// WMMA GEMM: logits[n, o] = sum_k h[n,k] * W[o,k] + b[o]
// One wave per 16x16 output tile, V_WMMA_F32_16X16X4_F32, K=128 in 32 steps.
// A (16x4 f32) layout: lane L -> M = L&15, holds K-pair at koff = 2*(L>>4).
// B (4x16 f32) layout mirrors A with N = L&15.
// C/D (16x16 f32): VGPR v -> M = v + 8*(L>>4), N = L&15.
// N is padded 40 -> 48: OOB columns read W row 0 (safe addr) with data zeroed,
// and their stores are masked.
// ---------------------------------------------------------------------------
__global__ __launch_bounds__(128) void sgc_gemm_wmma_kernel(
    const float* __restrict__ h,
    const float* __restrict__ W,
    const float* __restrict__ bias,
    float* __restrict__ out,
    int n_rows, int tilesN)
{
    int wave = (blockIdx.x * blockDim.x + threadIdx.x) >> 5;
    int lane = threadIdx.x & 31;
    int tileM = wave / tilesN;
    int tileN = wave - tileM * tilesN;

    int rowBase = tileM * 16;
    if (rowBase >= n_rows) return;                 // wave-uniform guard
    int colBase = tileN * 16;

    int half = lane >> 4;                          // 0: K 0..1, 1: K 2..3
    int m16  = lane & 15;
    int koff = half * 2;

    int col   = colBase + m16;
    bool colOK = col < F_OUT;
    int colClamped = colOK ? col : 0;

    const float* arow = h + (size_t)(rowBase + m16) * F_IN + koff;
    const float* brow = W + (size_t)colClamped * F_IN + koff;

    v8f acc = {};
    #pragma unroll
    for (int k = 0; k < F_IN; k += 4) {
        v2f a = *(const v2f*)(arow + k);
        v2f b = *(const v2f*)(brow + k);
        if (!colOK) { b.x = 0.0f; b.y = 0.0f; }
        // 8 args: (neg_a, A, neg_b, B, c_mod, C, reuse_a, reuse_b)
        acc = __builtin_amdgcn_wmma_f32_16x16x4_f32(
            false, a, false, b, (short)0, acc, false, false);
    }

    if (colOK) {
        float bv = bias[col];
        int rowOff = rowBase + half * 8;
        #pragma unroll
        for (int v = 0; v < 8; ++v) {
            out[(size_t)(rowOff + v) * F_OUT + col] = acc[v] + bv;
        }
    }
}

// ---------------------------------------------------------------------------
// Row-wise log-softmax over 40 columns, in place on d_out. One thread per row.
// ---------------------------------------------------------------------------
__global__ __launch_bounds__(256) void sgc_logsoftmax_kernel(
    float* __restrict__ out, int n_rows)
{
    int row = blockIdx.x * blockDim.x + threadIdx.x;
    if (row >= n_rows) return;
    float* p = out + (size_t)row * F_OUT;

    float v[F_OUT];
    float mx = -3.402823466e38f;
    #pragma unroll
    for (int j = 0; j < F_OUT; ++j) {
        v[j] = p[j];
        mx = fmaxf(mx, v[j]);
    }
    float s = 0.0f;
    #pragma unroll
    for (int j = 0; j < F_OUT; ++j) s += expf(v[j] - mx);
    float ls = logf(s) + mx;
    #pragma unroll
    for (int j = 0; j < F_OUT; ++j) p[j] = v[j] - ls;
}

// ---------------------------------------------------------------------------
// Launcher. Inputs (setup_inputs order):
//   0: x          [n_total*128] f32
//   1: edge_index [2*E]         i64   (src = first E, dst = second E)
//   2: edge_mask  [E]           i32
//   3: vertex_cnt (scalar, device)  -- unused on host; n_total derived from x
//   4: rule_cnt   (scalar, device)
//   5: W          [40*128]      f32
//   6: b          [40]          f32
// Workspace: h1, h2 buffers (n_total*128 f32 each), zeroed via memset node.
// ---------------------------------------------------------------------------
extern "C" void kernel_launch(void* const* d_in, const int* in_sizes, int n_in,
                              void* d_out, int out_size, void* d_ws, size_t ws_size,
                              hipStream_t stream) {
    const float*      x     = (const float*)d_in[0];
    const long long*  ei    = (const long long*)d_in[1];
    const int*        emask = (const int*)d_in[2];
    const float*      W     = (const float*)d_in[5];
    const float*      bias  = (const float*)d_in[6];
    float*            out   = (float*)d_out;

    int n_total = in_sizes[0] / F_IN;   // 120000
    int E       = in_sizes[1] / 2;      // 1.6M

    const long long* src = ei;
    const long long* dst = ei + E;

    size_t h_elems = (size_t)n_total * F_IN;
    float* h1 = (float*)d_ws;
    float* h2 = h1 + h_elems;

    hipMemsetAsync(d_ws, 0, 2 * h_elems * sizeof(float), stream);

    // Two propagation hops: x -> h1 -> h2
    int edgesPerBlock = 256 / 32;
    int hopGrid = (E + edgesPerBlock - 1) / edgesPerBlock;
    sgc_hop_kernel<<<hopGrid, 256, 0, stream>>>(x,  h1, src, dst, emask, E);
    sgc_hop_kernel<<<hopGrid, 256, 0, stream>>>(h1, h2, src, dst, emask, E);

    // WMMA GEMM: h2 @ W^T + b -> d_out (raw logits)
    int tilesM = (n_total + 15) / 16;          // 7500
    int tilesN = (F_OUT + 15) / 16;            // 3
    int waves  = tilesM * tilesN;
    int wavesPerBlock = 4;
    int gemmGrid = (waves + wavesPerBlock - 1) / wavesPerBlock;
    sgc_gemm_wmma_kernel<<<gemmGrid, wavesPerBlock * 32, 0, stream>>>(
        h2, W, bias, out, n_total, tilesN);

    // In-place log-softmax on d_out
    sgc_logsoftmax_kernel<<<(n_total + 255) / 256, 256, 0, stream>>>(out, n_total);
}